// DimwiseMedianConv_1906965479739
// MI455X (gfx1250) — compile-verified
//
#include <hip/hip_runtime.h>

// ---------------------------------------------------------------------------
// DimwiseMedianConv on gfx1250:
//   1) h = feat @ W via V_WMMA_F32_16X16X4_F32 (fp32 matrix pipe)
//   2) CSR-by-dst build (histogram + 8-elem/thread scan + scatter), self-loop w=1
//   3) weighted dimension-wise median per (node, dim): O(deg^2) selection over
//      LDS-packed (value,weight) pairs -> one ds_load_b64 per inner step
// ---------------------------------------------------------------------------

typedef __attribute__((ext_vector_type(2))) float v2f;
typedef __attribute__((ext_vector_type(8))) float v8f;

#define DIN  128
#define DOUT 64
#define MAXD 112   // LDS fast-path degree cap (observed max deg ~60 for this data)

// ------------------------- GEMM: h = feat @ W ------------------------------
// One wave computes a 16-row x 64-col tile; 8 waves (256 thr) per block.
// A(16x4 f32): lane<16 -> K={0,1}, lane>=16 -> K={2,3}; M = lane&15.
// B(4x16 f32): mirrored; C/D(16x16 f32): 8 VGPRs, M split across half-waves.
__global__ __launch_bounds__(256)
void gemm_wmma_f32(const float* __restrict__ feat, const float* __restrict__ W,
                   float* __restrict__ h, int n) {
  __shared__ float sW[DIN * DOUT];
  for (int i = threadIdx.x; i < DIN * DOUT; i += 256) sW[i] = W[i];
  __syncthreads();

  const int wave = threadIdx.x >> 5;
  const int lane = threadIdx.x & 31;
  int m0 = (blockIdx.x * 8 + wave) * 16;
  if (m0 + 16 > n) m0 = n - 16;            // uniform clamp: EXEC stays all-1s,
                                           // duplicate tiles write identical data
  const int row  = lane & 15;
  const int koff = (lane < 16) ? 0 : 2;
  const float* frow = feat + (size_t)(m0 + row) * DIN;

  v8f acc0 = {}, acc1 = {}, acc2 = {}, acc3 = {};
  for (int k = 0; k < DIN; k += 4) {
    v2f a;
    a.x = frow[k + koff];
    a.y = frow[k + koff + 1];
    const float* w0 = &sW[(k + koff)     * DOUT + row];
    const float* w1 = &sW[(k + koff + 1) * DOUT + row];
    v2f b0, b1, b2, b3;
    b0.x = w0[0];  b0.y = w1[0];
    b1.x = w0[16]; b1.y = w1[16];
    b2.x = w0[32]; b2.y = w1[32];
    b3.x = w0[48]; b3.y = w1[48];
    acc0 = __builtin_amdgcn_wmma_f32_16x16x4_f32(false, a, false, b0, (short)0, acc0, false, false);
    acc1 = __builtin_amdgcn_wmma_f32_16x16x4_f32(false, a, false, b1, (short)0, acc1, false, false);
    acc2 = __builtin_amdgcn_wmma_f32_16x16x4_f32(false, a, false, b2, (short)0, acc2, false, false);
    acc3 = __builtin_amdgcn_wmma_f32_16x16x4_f32(false, a, false, b3, (short)0, acc3, false, false);
  }

  const int rbase = (lane < 16) ? 0 : 8;
#pragma unroll
  for (int r = 0; r < 8; ++r) {
    float* o = h + (size_t)(m0 + rbase + r) * DOUT + row;
    o[0]  = acc0[r];
    o[16] = acc1[r];
    o[32] = acc2[r];
    o[48] = acc3[r];
  }
}

// --------------------- CSR build: init, count, scan, scatter ----------------
__global__ void init_kernel(int* __restrict__ deg, int* __restrict__ cur, int n) {
  int i = blockIdx.x * blockDim.x + threadIdx.x;
  if (i < n) { deg[i] = 1; cur[i] = 0; }   // deg starts at 1: the self-loop
}

__global__ void count_kernel(const int* __restrict__ ei, int* __restrict__ deg, int E) {
  int e = blockIdx.x * blockDim.x + threadIdx.x;
  if (e < E) atomicAdd(&deg[ei[E + e]], 1);   // ei[1][e] = dst
}

// Single-block exclusive scan, 8 elements per thread (8192/chunk) to minimize
// barrier count: ~7 chunks x 10 steps for n = 50000.
__global__ __launch_bounds__(1024)
void scan_kernel(const int* __restrict__ deg, int* __restrict__ rowstart, int n) {
  __shared__ int buf[1024];
  __shared__ int carry;
  if (threadIdx.x == 0) carry = 0;
  __syncthreads();
  for (int base = 0; base < n; base += 8192) {
    const int i0 = base + threadIdx.x * 8;
    int v[8];
    int s = 0;
#pragma unroll
    for (int t = 0; t < 8; ++t) {
      const int i = i0 + t;
      v[t] = (i < n) ? deg[i] : 0;
      s += v[t];
    }
    buf[threadIdx.x] = s;
    __syncthreads();
    for (int off = 1; off < 1024; off <<= 1) {
      int t = (threadIdx.x >= off) ? buf[threadIdx.x - off] : 0;
      __syncthreads();
      buf[threadIdx.x] += t;
      __syncthreads();
    }
    const int base_carry = carry;
    int running = base_carry + buf[threadIdx.x] - s;   // exclusive prefix of thread
#pragma unroll
    for (int t = 0; t < 8; ++t) {
      const int i = i0 + t;
      if (i < n) rowstart[i] = running;
      running += v[t];
    }
    __syncthreads();
    if (threadIdx.x == 1023) carry = base_carry + buf[1023];
    __syncthreads();
  }
  if (threadIdx.x == 0) rowstart[n] = carry;
}

__global__ void scatter_self(const int* __restrict__ rowstart, int* __restrict__ cur,
                             int* __restrict__ csr_src, float* __restrict__ csr_w, int n) {
  int i = blockIdx.x * blockDim.x + threadIdx.x;
  if (i >= n) return;
  int p = rowstart[i] + atomicAdd(&cur[i], 1);
  csr_src[p] = i;
  csr_w[p]   = 1.0f;
}

__global__ void scatter_edges(const int* __restrict__ ei, const float* __restrict__ ew,
                              const int* __restrict__ rowstart, int* __restrict__ cur,
                              int* __restrict__ csr_src, float* __restrict__ csr_w, int E) {
  int e = blockIdx.x * blockDim.x + threadIdx.x;
  if (e >= E) return;
  const int dst = ei[E + e];
  int p = rowstart[dst] + atomicAdd(&cur[dst], 1);
  csr_src[p] = ei[e];
  csr_w[p]   = ew[e];
}

// --------------------- weighted dim-wise median ----------------------------
// One block per dst node, one thread per output dim (64 thr = 2 waves).
// (value, weight) packed per (edge, dim) in LDS: the O(deg^2) hot loop does a
// single ds_load_b64 per step; 32 lanes x 2 dwords exactly cover the 64 LDS
// banks -> conflict-free. Answer = min (value, pos) among candidates whose
// inclusive stable-order cumulative weight >= half (the reference's
// first-crossing element).
struct __align__(8) VW { float v; float w; };

__global__ __launch_bounds__(DOUT)
void median_kernel(const float* __restrict__ h, const int* __restrict__ rowstart,
                   const int* __restrict__ csr_src, const float* __restrict__ csr_w,
                   const float* __restrict__ bias, float* __restrict__ out) {
  const int node = blockIdx.x;
  const int d    = threadIdx.x;
  const int beg  = rowstart[node];
  const int deg  = rowstart[node + 1] - beg;

  __shared__ float sw[MAXD];
  __shared__ int   ssrc[MAXD];
  __shared__ VW    spairs[MAXD * DOUT];

  float ans;
  if (deg <= MAXD) {
    for (int j = d; j < deg; j += DOUT) {
      ssrc[j] = csr_src[beg + j];
      sw[j]   = csr_w[beg + j];
    }
    __syncthreads();
    float tot = 0.f;
    for (int j = 0; j < deg; ++j) {
      VW p;
      p.v = h[(size_t)ssrc[j] * DOUT + d];   // coalesced: 64 consecutive floats
      p.w = sw[j];
      spairs[j * DOUT + d] = p;
      tot += p.w;
    }
    __syncthreads();
    const float half = 0.5f * tot;

    float bestv = __builtin_inff();
    int   bestp = 0x7fffffff;
    for (int i = 0; i < deg; ++i) {
      const float vi = spairs[i * DOUT + d].v;
      float cw = 0.f;
      for (int j = 0; j < deg; ++j) {
        const VW p = spairs[j * DOUT + d];
        const bool le = (p.v < vi) || (p.v == vi && j <= i);
        cw += le ? p.w : 0.f;
      }
      if (cw >= half && (vi < bestv || (vi == bestv && i < bestp))) {
        bestv = vi; bestp = i;
      }
    }
    ans = bestv;
  } else {
    // slow path (never taken for this data's degree distribution)
    float tot = 0.f;
    for (int j = 0; j < deg; ++j) tot += csr_w[beg + j];
    const float half = 0.5f * tot;
    float bestv = __builtin_inff();
    int   bestp = 0x7fffffff;
    for (int i = 0; i < deg; ++i) {
      const float vi = h[(size_t)csr_src[beg + i] * DOUT + d];
      float cw = 0.f;
      for (int j = 0; j < deg; ++j) {
        const float vj = h[(size_t)csr_src[beg + j] * DOUT + d];
        const bool le = (vj < vi) || (vj == vi && j <= i);
        cw += le ? csr_w[beg + j] : 0.f;
      }
      if (cw >= half && (vi < bestv || (vi == bestv && i < bestp))) {
        bestv = vi; bestp = i;
      }
    }
    ans = bestv;
  }
  out[(size_t)node * DOUT + d] = ans + bias[d];
}

// --------------------------------- launch ----------------------------------
extern "C" void kernel_launch(void* const* d_in, const int* in_sizes, int n_in,
                              void* d_out, int out_size, void* d_ws, size_t ws_size,
                              hipStream_t stream) {
  const float* feat = (const float*)d_in[0];
  const int*   ei   = (const int*)  d_in[1];
  const float* ew   = (const float*)d_in[2];
  const float* W    = (const float*)d_in[3];
  const float* bias = (const float*)d_in[4];
  float* out = (float*)d_out;

  const int E    = in_sizes[2];        // edge_weight count
  const int n    = in_sizes[0] / DIN;  // nodes
  const int Etot = E + n;              // edges + self loops
  (void)n_in; (void)out_size; (void)ws_size;

  // workspace carve-out (~26.6 MB total)
  char* ws = (char*)d_ws;
  size_t off = 0;
  auto carve = [&](size_t bytes) {
    char* p = ws + off;
    off = (off + bytes + 255) & ~(size_t)255;
    return p;
  };
  float* h        = (float*)carve((size_t)n * DOUT * 4);
  int*   deg      = (int*)  carve((size_t)n * 4);
  int*   rowstart = (int*)  carve((size_t)(n + 1) * 4);
  int*   cur      = (int*)  carve((size_t)n * 4);
  int*   csr_src  = (int*)  carve((size_t)Etot * 4);
  float* csr_w    = (float*)carve((size_t)Etot * 4);

  // 1) GEMM on the matrix pipe
  const int tiles = (n + 15) / 16;
  gemm_wmma_f32<<<(tiles + 7) / 8, 256, 0, stream>>>(feat, W, h, n);

  // 2) CSR by destination
  init_kernel   <<<(n + 255) / 256, 256, 0, stream>>>(deg, cur, n);
  count_kernel  <<<(E + 255) / 256, 256, 0, stream>>>(ei, deg, E);
  scan_kernel   <<<1, 1024, 0, stream>>>(deg, rowstart, n);
  scatter_self  <<<(n + 255) / 256, 256, 0, stream>>>(rowstart, cur, csr_src, csr_w, n);
  scatter_edges <<<(E + 255) / 256, 256, 0, stream>>>(ei, ew, rowstart, cur, csr_src, csr_w, E);

  // 3) weighted dimension-wise median + bias
  median_kernel<<<n, DOUT, 0, stream>>>(h, rowstart, csr_src, csr_w, bias, out);
}